// StockBlockLayer_11458972745770
// MI455X (gfx1250) — compile-verified
//
#include <hip/hip_runtime.h>
#include <math.h>

// ---------------------------------------------------------------------------
// MI455X (gfx1250). All matmuls on V_WMMA_F32_16X16X4_F32 (fp32 reference,
// L2-resident problem -> fp32 matrix path, no precision loss).
//
// WMMA f32 16x16x4 per-lane layout (CDNA5 ISA 7.12.2):
//   lane = tid&31, col = lane&15, half = lane>>4, k2 = 2*half
//   A (16x4):  a.x = A[col][k2],  a.y = A[col][k2+1]
//   B (4x16):  b.x = B[k2][col],  b.y = B[k2+1][col]
//   C/D 16x16: acc[j] = C[j + 8*half][col]
// All B operands are staged K-major so each lane's (B[k],B[k+1]) pair is
// contiguous -> one 64-bit load, no v_mov packing in the WMMA loops.
// ---------------------------------------------------------------------------

typedef __attribute__((ext_vector_type(2))) float v2f;
typedef __attribute__((ext_vector_type(8))) float v8f;

__device__ __forceinline__ v8f wmma4(v2f a, v2f b, v8f c) {
  return __builtin_amdgcn_wmma_f32_16x16x4_f32(false, a, false, b, (short)0, c,
                                               false, false);
}

#define NB   32     // batch
#define NN   512    // nodes
#define NT   64     // T
#define NK   4      // ORDER
#define NL   128    // L
#define NM   16     // MODES

// ---------------------------------------------------------------------------
// K0: init. DFT basis tables (transposed, K-major for WMMA B operands) and
// a transposed copy of `weight`.
//  basis[0:2048)     FcT[m][l] =  cos(2*pi*index[m]*l/128)      (m*128+l)
//  basis[2048:4096)  FsT[m][l] = -sin(2*pi*index[m]*l/128)
//  basis[4096:6144)  CrT[l][m] =  c_m*cos(2*pi*m*l/128)         (l*16+m)
//  basis[6144:8192)  CiT[l][m] = -c_m*sin(2*pi*m*l/128); CiT[l][0]=0
//                    (irfft ignores Im of bin 0);  c_0=1/128, c_m=2/128
//  wT[k][l][t] = weight[k][t][l]
// ---------------------------------------------------------------------------
__global__ __launch_bounds__(256) void k_init(const int* __restrict__ index,
                                              const float* __restrict__ weight,
                                              float* __restrict__ basis,
                                              float* __restrict__ wT) {
  const int tid = threadIdx.x;
  const float w0 = 6.283185307179586f / 128.0f;
#pragma unroll
  for (int i = 0; i < 8; ++i) {
    const int idx = i * 256 + tid;          // 0..2047 -> (m, l)
    const int m = idx >> 7, l = idx & 127;
    const int ph = (index[m] * l) & 127;    // period 128
    float s, c;
    __sincosf(w0 * (float)ph, &s, &c);
    basis[idx]        =  c;                 // FcT
    basis[2048 + idx] = -s;                 // FsT
  }
#pragma unroll
  for (int i = 0; i < 8; ++i) {
    const int idx = i * 256 + tid;          // 0..2047 -> (l, m)
    const int l = idx >> 4, m = idx & 15;
    const int ph = (m * l) & 127;
    float s, c;
    __sincosf(w0 * (float)ph, &s, &c);
    const float cm = (m == 0) ? (1.0f / 128.0f) : (2.0f / 128.0f);
    basis[4096 + idx] = cm * c;                         // CrT
    basis[6144 + idx] = (m == 0) ? 0.0f : (-cm * s);    // CiT
  }
  // transpose weight: wT[k][l][t]
  for (int i = 0; i < 128; ++i) {
    const int idx = i * 256 + tid;          // 0..32767
    const int k = idx >> 13, r = idx & 8191;
    const int l = r >> 6, t = r & 63;
    wT[idx] = weight[(size_t)k * (NT * NL) + (size_t)t * NL + l];
  }
}

// ---------------------------------------------------------------------------
// K1: tmp[b,k,n,t] = sum_m mul_L[k,n,m] * x[b,m,t]
// grid (N/128, B, K); block 256 = 8 waves (4x2); 128x64 block tile,
// 32x32 per wave (4 accumulators); K-chunks of 32.
// ---------------------------------------------------------------------------
__global__ __launch_bounds__(256) void k_gemm1(const float* __restrict__ mulL,
                                               const float* __restrict__ x,
                                               float* __restrict__ tmp) {
  const int n0 = blockIdx.x * 128;
  const int b  = blockIdx.y;
  const int k  = blockIdx.z;

  __shared__ __align__(16) float As [128][34];  // [n][m], padded
  __shared__ __align__(16) float BsT[64][34];   // [t][m], K-major pairs

  const int tid  = threadIdx.x;
  const int wave = tid >> 5, lane = tid & 31;
  const int wr = wave >> 1, wc = wave & 1;          // 4x2 wave grid
  const int col = lane & 15, half = lane >> 4, k2 = half * 2;

  v8f acc00 = {}, acc01 = {}, acc10 = {}, acc11 = {};

  const float* Ag = mulL + (size_t)k * (NN * NN) + (size_t)n0 * NN;
  const float* Bg = x + (size_t)b * (NN * NT);

  const int ra = tid >> 1, ca = (tid & 1) * 16;     // A: 128 rows x 32 cols
  const int rm = tid >> 3, ct = (tid & 7) * 8;      // B: 32 m-rows x 64 t-cols

  for (int m0 = 0; m0 < NN; m0 += 32) {
    {
      const float* s0 = Ag + (size_t)ra * NN + m0 + ca;
      float4 a0 = *(const float4*)(s0);
      float4 a1 = *(const float4*)(s0 + 4);
      float4 a2 = *(const float4*)(s0 + 8);
      float4 a3 = *(const float4*)(s0 + 12);
      float va[16] = {a0.x, a0.y, a0.z, a0.w, a1.x, a1.y, a1.z, a1.w,
                      a2.x, a2.y, a2.z, a2.w, a3.x, a3.y, a3.z, a3.w};
#pragma unroll
      for (int q = 0; q < 16; ++q) As[ra][ca + q] = va[q];

      const float* s1 = Bg + (size_t)(m0 + rm) * NT + ct;
      float4 b0 = *(const float4*)(s1);
      float4 b1 = *(const float4*)(s1 + 4);
      float vb[8] = {b0.x, b0.y, b0.z, b0.w, b1.x, b1.y, b1.z, b1.w};
#pragma unroll
      for (int q = 0; q < 8; ++q) BsT[ct + q][rm] = vb[q];  // transpose
    }
    if (m0 + 32 < NN) {  // prefetch next chunk -> global_prefetch_b8
      __builtin_prefetch(Ag + (size_t)ra * NN + m0 + 32 + ca, 0, 3);
      __builtin_prefetch(Bg + (size_t)(m0 + 32 + rm) * NT + ct, 0, 3);
    }
    __syncthreads();
#pragma unroll
    for (int kk = 0; kk < 32; kk += 4) {
      v2f a0; a0.x = As[wr * 32 + col][kk + k2];
              a0.y = As[wr * 32 + col][kk + k2 + 1];
      v2f a1; a1.x = As[wr * 32 + 16 + col][kk + k2];
              a1.y = As[wr * 32 + 16 + col][kk + k2 + 1];
      v2f b0; b0.x = BsT[wc * 32 + col][kk + k2];
              b0.y = BsT[wc * 32 + col][kk + k2 + 1];
      v2f b1; b1.x = BsT[wc * 32 + 16 + col][kk + k2];
              b1.y = BsT[wc * 32 + 16 + col][kk + k2 + 1];
      acc00 = wmma4(a0, b0, acc00);
      acc01 = wmma4(a0, b1, acc01);
      acc10 = wmma4(a1, b0, acc10);
      acc11 = wmma4(a1, b1, acc11);
    }
    __syncthreads();
  }

  float* out =
      tmp + (((size_t)b * NK + k) * NN + n0 + wr * 32) * NT + wc * 32;
#pragma unroll
  for (int j = 0; j < 8; ++j) {
    const int row = j + 8 * half;
    out[(size_t)row * NT + col]             = acc00[j];
    out[(size_t)row * NT + 16 + col]        = acc01[j];
    out[(size_t)(row + 16) * NT + col]      = acc10[j];
    out[(size_t)(row + 16) * NT + 16 + col] = acc11[j];
  }
}

// ---------------------------------------------------------------------------
// K2: fused  g = tmp @ weight[k];  sel = g @ {Fc,Fs};  res = complex 4x4 mix;
//            four = resR@Cr + resI@Ci;  igf = sum_k (g + four)
// (series_decomp is the identity: softmax over a size-1 axis == 1, so
//  x3 == x2 exactly in real arithmetic; dec_w/dec_b are unused.)
// grid (N/16, B); block 256 = 8 waves.
// ---------------------------------------------------------------------------
__global__ __launch_bounds__(256) void k_fourier(const float* __restrict__ tmp,
                                                 const float* __restrict__ wT,
                                                 const float* __restrict__ w1r,
                                                 const float* __restrict__ w1i,
                                                 const float* __restrict__ basis,
                                                 float* __restrict__ igf) {
  const int n0 = blockIdx.x * 16;
  const int b  = blockIdx.y;

  __shared__ __align__(16) float Ts[NK][16][66];   // tmp tile  [k][n][t]
  __shared__ __align__(16) float Gs[NK][16][130];  // g then x2 [k][n][l]
  __shared__ float S0[NK][16][18];                 // selR then resR
  __shared__ float S1[NK][16][18];                 // selI then resI

  const int tid  = threadIdx.x;
  const int wave = tid >> 5, lane = tid & 31;
  const int col = lane & 15, half = lane >> 4, k2 = half * 2;

  const float* FcT = basis;          // [m][l]
  const float* FsT = basis + 2048;
  const float* CrT = basis + 4096;   // [l][m]
  const float* CiT = basis + 6144;

  // ---- stage 1: load tmp tile [4][16][64] ----
  {
    const int r = tid >> 2;                 // 0..63
    const int kq = r >> 4, nq = r & 15;
    const int c0 = (tid & 3) * 16;
    const float* src =
        tmp + (((size_t)b * NK + kq) * NN + n0 + nq) * NT + c0;
    float4 v0 = *(const float4*)(src);
    float4 v1 = *(const float4*)(src + 4);
    float4 v2 = *(const float4*)(src + 8);
    float4 v3 = *(const float4*)(src + 12);
    float vv[16] = {v0.x, v0.y, v0.z, v0.w, v1.x, v1.y, v1.z, v1.w,
                    v2.x, v2.y, v2.z, v2.w, v3.x, v3.y, v3.z, v3.w};
#pragma unroll
    for (int q = 0; q < 16; ++q) Ts[kq][nq][c0 + q] = vv[q];
  }
  __syncthreads();

  // ---- stage 2: g[k] = Ts[k] (16x64) @ weight[k] (64x128), B from wT ----
#pragma unroll
  for (int i = 0; i < 4; ++i) {
    const int t  = wave * 4 + i;            // 32 tiles: k x 8 l-tiles
    const int kq = t >> 3, lt = t & 7;
    const float* W = wT + (size_t)kq * (NL * NT) + (size_t)(lt * 16 + col) * NT;
    v8f acc = {};
#pragma unroll
    for (int kk = 0; kk < 64; kk += 4) {
      v2f a;  a.x = Ts[kq][col][kk + k2];  a.y = Ts[kq][col][kk + k2 + 1];
      v2f bb; bb.x = W[kk + k2];           bb.y = W[kk + k2 + 1];
      acc = wmma4(a, bb, acc);
    }
#pragma unroll
    for (int j = 0; j < 8; ++j) Gs[kq][j + 8 * half][lt * 16 + col] = acc[j];
  }
  __syncthreads();

  // ---- stage 3: sel = g (16x128) @ Fc/Fs (128x16); one tile per wave ----
  {
    const int kq = wave >> 1, ri = wave & 1;
    const float* F = (ri ? FsT : FcT) + (size_t)col * NL;
    v8f acc = {};
#pragma unroll
    for (int kk = 0; kk < 128; kk += 4) {
      v2f a;  a.x = Gs[kq][col][kk + k2];  a.y = Gs[kq][col][kk + k2 + 1];
      v2f bb; bb.x = F[kk + k2];           bb.y = F[kk + k2 + 1];
      acc = wmma4(a, bb, acc);
    }
    float(*S)[16][18] = ri ? S1 : S0;
#pragma unroll
    for (int j = 0; j < 8; ++j) S[kq][j + 8 * half][col] = acc[j];
  }
  __syncthreads();

  // ---- stage 4: res[o] = sum_i sel[i] * w1[n,i,o,m]  (complex, VALU) ----
  float rr[4], ii[4];
#pragma unroll
  for (int q = 0; q < 4; ++q) {
    const int idx = q * 256 + tid;          // 0..1023 -> (o, n, m)
    const int o = idx >> 8, nq = (idx >> 4) & 15, m = idx & 15;
    const size_t wb = (size_t)(n0 + nq) * 256 + (size_t)o * 16 + m;  // + i*64
    float ar = 0.f, ai = 0.f;
#pragma unroll
    for (int i2 = 0; i2 < 4; ++i2) {
      const float wr_ = w1r[wb + (size_t)i2 * 64];
      const float wi_ = w1i[wb + (size_t)i2 * 64];
      const float sr = S0[i2][nq][m], si = S1[i2][nq][m];
      ar += sr * wr_ - si * wi_;
      ai += sr * wi_ + si * wr_;
    }
    rr[q] = ar; ii[q] = ai;
  }
  __syncthreads();   // all sel reads done; reuse S0/S1 for res
#pragma unroll
  for (int q = 0; q < 4; ++q) {
    const int idx = q * 256 + tid;
    const int o = idx >> 8, nq = (idx >> 4) & 15, m = idx & 15;
    S0[o][nq][m] = rr[q];
    S1[o][nq][m] = ii[q];
  }
  __syncthreads();

  // ---- stage 5: four = resR@Cr + resI@Ci, accumulated into Gs (x2=g+four) --
#pragma unroll
  for (int i = 0; i < 4; ++i) {
    const int t = wave * 4 + i;
    const int o = t >> 3, lt = t & 7;
    const float* Cr = CrT + (size_t)(lt * 16 + col) * NM;
    const float* Ci = CiT + (size_t)(lt * 16 + col) * NM;
    v8f acc = {};
#pragma unroll
    for (int kk = 0; kk < 16; kk += 4) {
      v2f a;  a.x = S0[o][col][kk + k2];   a.y = S0[o][col][kk + k2 + 1];
      v2f bb; bb.x = Cr[kk + k2];          bb.y = Cr[kk + k2 + 1];
      acc = wmma4(a, bb, acc);
      v2f a2;  a2.x = S1[o][col][kk + k2];  a2.y = S1[o][col][kk + k2 + 1];
      v2f b2;  b2.x = Ci[kk + k2];          b2.y = Ci[kk + k2 + 1];
      acc = wmma4(a2, b2, acc);
    }
#pragma unroll
    for (int j = 0; j < 8; ++j) Gs[o][j + 8 * half][lt * 16 + col] += acc[j];
  }
  __syncthreads();

  // ---- stage 6: igf[b,n,l] = sum_k x2 ----
#pragma unroll
  for (int q = 0; q < 8; ++q) {
    const int idx = q * 256 + tid;          // 0..2047
    const int nq = idx >> 7, l = idx & 127;
    const float s = Gs[0][nq][l] + Gs[1][nq][l] + Gs[2][nq][l] + Gs[3][nq][l];
    igf[((size_t)b * NN + n0 + nq) * NL + l] = s;
  }
}

// ---------------------------------------------------------------------------
// K3: tail.  rows = flattened [B*N]; 64 rows per block.
//   fsrc = igf@fc_w^T + fc_b ; forecast = fsrc@fr_w^T + fr_b
//   bc = igf@bc_w^T + bc_b ; bshort = x@bs_w^T + bs_b
//   out = [forecast | sigmoid(bshort-bc) | sigmoid(bc)]
// B operands (fc_w[j*128+k], +1) are already K-adjacent in global memory and
// L2-resident (144 KB total shared by all blocks).
// ---------------------------------------------------------------------------
__global__ __launch_bounds__(256) void k_tail(
    const float* __restrict__ igf, const float* __restrict__ x,
    const float* __restrict__ fc_w, const float* __restrict__ fc_b,
    const float* __restrict__ fr_w, const float* __restrict__ fr_b,
    const float* __restrict__ bc_w, const float* __restrict__ bc_b,
    const float* __restrict__ bs_w, const float* __restrict__ bs_b,
    float* __restrict__ out) {
  const int r0 = blockIdx.x * 64;

  __shared__ __align__(16) float Fsrc[64][130];

  const int tid  = threadIdx.x;
  const int wave = tid >> 5, lane = tid & 31;
  const int wr = wave >> 1, wc = wave & 1;
  const int col = lane & 15, half = lane >> 4, k2 = half * 2;

  const float* Arow = igf + (size_t)(r0 + wr * 16) * NL;  // this wave's A rows

  // fsrc (64x128): wave -> rows wr*16..+16, cols wc*64 + jt*16
#pragma unroll
  for (int jt = 0; jt < 4; ++jt) {
    const int cb = wc * 64 + jt * 16;
    const float* Bw = fc_w + (size_t)(cb + col) * NL;
    v8f acc = {};
#pragma unroll
    for (int kk = 0; kk < 128; kk += 4) {
      v2f a;  a.x = Arow[(size_t)col * NL + kk + k2];
              a.y = Arow[(size_t)col * NL + kk + k2 + 1];
      v2f bb; bb.x = Bw[kk + k2];  bb.y = Bw[kk + k2 + 1];
      acc = wmma4(a, bb, acc);
    }
    const float bias = fc_b[cb + col];
#pragma unroll
    for (int j = 0; j < 8; ++j)
      Fsrc[wr * 16 + j + 8 * half][cb + col] = acc[j] + bias;
  }
  __syncthreads();

  // forecast (64x64)
#pragma unroll
  for (int jt = 0; jt < 2; ++jt) {
    const int cb = wc * 32 + jt * 16;
    const float* Bw = fr_w + (size_t)(cb + col) * NL;
    v8f acc = {};
#pragma unroll
    for (int kk = 0; kk < 128; kk += 4) {
      v2f a;  a.x = Fsrc[wr * 16 + col][kk + k2];
              a.y = Fsrc[wr * 16 + col][kk + k2 + 1];
      v2f bb; bb.x = Bw[kk + k2];  bb.y = Bw[kk + k2 + 1];
      acc = wmma4(a, bb, acc);
    }
    const float bias = fr_b[cb + col];
#pragma unroll
    for (int j = 0; j < 8; ++j) {
      const int row = r0 + wr * 16 + j + 8 * half;
      out[(size_t)row * NT + cb + col] = acc[j] + bias;
    }
  }

  // bc (igf@bc_w^T) and bshort (x@bs_w^T) -> sigmoids
#pragma unroll
  for (int jt = 0; jt < 2; ++jt) {
    const int cb = wc * 32 + jt * 16;
    const float* Bw = bc_w + (size_t)(cb + col) * NL;
    v8f accB = {};
#pragma unroll
    for (int kk = 0; kk < 128; kk += 4) {
      v2f a;  a.x = Arow[(size_t)col * NL + kk + k2];
              a.y = Arow[(size_t)col * NL + kk + k2 + 1];
      v2f bb; bb.x = Bw[kk + k2];  bb.y = Bw[kk + k2 + 1];
      accB = wmma4(a, bb, accB);
    }
    const float* Bs2 = bs_w + (size_t)(cb + col) * NT;
    v8f accS = {};
#pragma unroll
    for (int kk = 0; kk < 64; kk += 4) {
      v2f a;  a.x = x[(size_t)(r0 + wr * 16 + col) * NT + kk + k2];
              a.y = x[(size_t)(r0 + wr * 16 + col) * NT + kk + k2 + 1];
      v2f bb; bb.x = Bs2[kk + k2];  bb.y = Bs2[kk + k2 + 1];
      accS = wmma4(a, bb, accS);
    }
    const float biasB = bc_b[cb + col];
    const float biasS = bs_b[cb + col];
#pragma unroll
    for (int j = 0; j < 8; ++j) {
      const int row = r0 + wr * 16 + j + 8 * half;
      const float bcv = accB[j] + biasB;
      const float bsv = accS[j] + biasS;
      const size_t oi = (size_t)row * NT + cb + col;
      out[1048576 + oi] = 1.0f / (1.0f + __expf(-(bsv - bcv)));  // backcast_src
      out[2097152 + oi] = 1.0f / (1.0f + __expf(-bcv));          // x_back
    }
  }
}

// ---------------------------------------------------------------------------
extern "C" void kernel_launch(void* const* d_in, const int* in_sizes, int n_in,
                              void* d_out, int out_size, void* d_ws,
                              size_t ws_size, hipStream_t stream) {
  (void)in_sizes; (void)n_in; (void)out_size; (void)ws_size;

  const float* x      = (const float*)d_in[0];
  const float* mul_L  = (const float*)d_in[1];
  const float* weight = (const float*)d_in[2];
  const float* w1r    = (const float*)d_in[3];
  const float* w1i    = (const float*)d_in[4];
  const float* fc_w   = (const float*)d_in[5];
  const float* fc_b   = (const float*)d_in[6];
  const float* fr_w   = (const float*)d_in[7];
  const float* fr_b   = (const float*)d_in[8];
  const float* bc_w   = (const float*)d_in[9];
  const float* bc_b   = (const float*)d_in[10];
  const float* bs_w   = (const float*)d_in[11];
  const float* bs_b   = (const float*)d_in[12];
  // d_in[13] dec_w, d_in[14] dec_b: provably unused (softmax over size-1 axis)
  const int*   index  = (const int*)d_in[15];

  float* out   = (float*)d_out;
  float* tmp   = (float*)d_ws;                       // 32*4*512*64   = 16 MB
  float* igf   = tmp + (size_t)NB * NK * NN * NT;    // 32*512*128    =  8 MB
  float* basis = igf + (size_t)NB * NN * NL;         // 8192 floats   = 32 KB
  float* wT    = basis + 8192;                       // 32768 floats  = 128 KB

  k_init   <<<1, 256, 0, stream>>>(index, weight, basis, wT);
  k_gemm1  <<<dim3(NN / 128, NB, NK), 256, 0, stream>>>(mul_L, x, tmp);
  k_fourier<<<dim3(NN / 16, NB), 256, 0, stream>>>(tmp, wT, w1r, w1i,
                                                   basis, igf);
  k_tail   <<<dim3((NB * NN) / 64), 256, 0, stream>>>(
      igf, x, fc_w, fc_b, fr_w, fr_b, bc_w, bc_b, bs_w, bs_b, out);
}